// HybridRBF_65481071403321
// MI455X (gfx1250) — compile-verified
//
#include <hip/hip_runtime.h>
#include <math.h>

#ifndef __has_builtin
#define __has_builtin(x) 0
#endif

#define THREADS         256
#define ROWS_PER_BLOCK  128
#define DIM             64
#define ROW_STRIDE      68   // 64 + 4 dword pad: rotates LDS banks per row

// ---------------------------------------------------------------------------
// CDNA5 async global->LDS copy (GLOBAL_LOAD_ASYNC_TO_LDS_B128, ASYNCcnt).
// ROCm 7.2 clang types the b128 builtin as (int4 global*, int4 local*, Ii, Ii).
// ---------------------------------------------------------------------------
typedef int v4i __attribute__((vector_size(16)));
typedef __attribute__((address_space(1))) v4i* g_v4i_p;
typedef __attribute__((address_space(3))) v4i* l_v4i_p;

__device__ __forceinline__ void async_copy_b128(const void* gsrc, void* ldst) {
#if __has_builtin(__builtin_amdgcn_global_load_async_to_lds_b128)
    __builtin_amdgcn_global_load_async_to_lds_b128(
        (g_v4i_p)(gsrc), (l_v4i_p)(ldst), 0, 0);
#else
    unsigned lds_off = (unsigned)(unsigned long long)
        (__attribute__((address_space(3))) void*)(ldst);
    asm volatile("global_load_async_to_lds_b128 %0, %1, off"
                 :: "v"(lds_off), "v"(gsrc) : "memory");
#endif
}

__device__ __forceinline__ void wait_async0() {
#if __has_builtin(__builtin_amdgcn_s_wait_asynccnt)
    __builtin_amdgcn_s_wait_asynccnt(0);
#else
    asm volatile("s_wait_asynccnt 0" ::: "memory");
#endif
}

__device__ __forceinline__ void acc4(const float4 xv, const float4 yv,
                                     float& sx, float& sy,
                                     float& sxx, float& syy, float& sxy) {
    sx += xv.x + xv.y + xv.z + xv.w;
    sy += yv.x + yv.y + yv.z + yv.w;
    sxx = fmaf(xv.x, xv.x, sxx); sxx = fmaf(xv.y, xv.y, sxx);
    sxx = fmaf(xv.z, xv.z, sxx); sxx = fmaf(xv.w, xv.w, sxx);
    syy = fmaf(yv.x, yv.x, syy); syy = fmaf(yv.y, yv.y, syy);
    syy = fmaf(yv.z, yv.z, syy); syy = fmaf(yv.w, yv.w, syy);
    sxy = fmaf(xv.x, yv.x, sxy); sxy = fmaf(xv.y, yv.y, sxy);
    sxy = fmaf(xv.z, yv.z, sxy); sxy = fmaf(xv.w, yv.w, sxy);
}

__global__ __launch_bounds__(THREADS)
void hybrid_rbf_kernel(const float* __restrict__ x,
                       const float* __restrict__ y,
                       float* __restrict__ out,
                       int n_rows) {
    __shared__ __align__(16) float xs[ROWS_PER_BLOCK * ROW_STRIDE];
    __shared__ __align__(16) float ys[ROWS_PER_BLOCK * ROW_STRIDE];

    const int tid = threadIdx.x;
    const long long row_base = (long long)blockIdx.x * ROWS_PER_BLOCK;
    const bool full_tile = (row_base + ROWS_PER_BLOCK) <= (long long)n_rows;

    if (full_tile) {
        // Tile is a contiguous 32KB block per tensor: 2048 16B chunks each,
        // 8 chunks/thread/tensor -> 16 outstanding async b128 loads per lane.
        const float* gx = x + row_base * DIM;
        const float* gy = y + row_base * DIM;
#pragma unroll
        for (int k = 0; k < 8; ++k) {
            int c   = tid + k * THREADS;       // chunk id 0..2047
            int r   = c >> 4;                  // row in tile (16 chunks/row)
            int col = c & 15;                  // 16B chunk within row
            int lo  = r * ROW_STRIDE + col * 4;
            async_copy_b128(gx + c * 4, &xs[lo]);
            async_copy_b128(gy + c * 4, &ys[lo]);
        }
        wait_async0();
    } else {
        // Partial last tile: guarded synchronous fill (zeros for OOB rows).
#pragma unroll
        for (int k = 0; k < 8; ++k) {
            int c   = tid + k * THREADS;
            int r   = c >> 4;
            int col = c & 15;
            long long grow = row_base + r;
            float4 vx = make_float4(0.f, 0.f, 0.f, 0.f);
            float4 vy = make_float4(0.f, 0.f, 0.f, 0.f);
            if (grow < (long long)n_rows) {
                vx = *(const float4*)(x + grow * DIM + col * 4);
                vy = *(const float4*)(y + grow * DIM + col * 4);
            }
            *(float4*)&xs[r * ROW_STRIDE + col * 4] = vx;
            *(float4*)&ys[r * ROW_STRIDE + col * 4] = vy;
        }
    }
    __syncthreads();

    const int lane = tid & 31;
    const int wave = tid >> 5;     // 0..7
    const int g    = lane >> 2;    // row slot within wave: 0..7
    const int q    = lane & 3;     // quarter of row: dwords q*16 .. q*16+15

#pragma unroll
    for (int pass = 0; pass < ROWS_PER_BLOCK / 64; ++pass) {
        const int lr = pass * 64 + wave * 8 + g;    // local row 0..127
        const float4* xr = (const float4*)&xs[lr * ROW_STRIDE + q * 16];
        const float4* yr = (const float4*)&ys[lr * ROW_STRIDE + q * 16];
        float4 x0 = xr[0], x1 = xr[1], x2 = xr[2], x3 = xr[3];
        float4 y0 = yr[0], y1 = yr[1], y2 = yr[2], y3 = yr[3];

        float sx = 0.f, sy = 0.f, sxx = 0.f, syy = 0.f, sxy = 0.f;
        acc4(x0, y0, sx, sy, sxx, syy, sxy);
        acc4(x1, y1, sx, sy, sxx, syy, sxy);
        acc4(x2, y2, sx, sy, sxx, syy, sxy);
        acc4(x3, y3, sx, sy, sxx, syy, sxy);

        // quantum term: product of cos(x_d - y_d) for d < 8; only lane q==0
        // holds d=0..15, and only it writes the result, so no reduction needed.
        float cp = 1.0f;
        if (q == 0) {
            cp = __cosf(x0.x - y0.x) * __cosf(x0.y - y0.y) *
                 __cosf(x0.z - y0.z) * __cosf(x0.w - y0.w) *
                 __cosf(x1.x - y1.x) * __cosf(x1.y - y1.y) *
                 __cosf(x1.z - y1.z) * __cosf(x1.w - y1.w);
        }

        // reduce the 5 moments across the 4 lanes of this row (xor 1,2 stays
        // inside the 4-lane group on wave32)
        sx  += __shfl_xor(sx, 1, 32);
        sy  += __shfl_xor(sy, 1, 32);
        sxx += __shfl_xor(sxx, 1, 32);
        syy += __shfl_xor(syy, 1, 32);
        sxy += __shfl_xor(sxy, 1, 32);
        sx  += __shfl_xor(sx, 2, 32);
        sy  += __shfl_xor(sy, 2, 32);
        sxx += __shfl_xor(sxx, 2, 32);
        syy += __shfl_xor(syy, 2, 32);
        sxy += __shfl_xor(sxy, 2, 32);

        if (q == 0) {
            long long grow = row_base + lr;
            if (grow < (long long)n_rows) {
                const float mx = sx * (1.0f / 64.0f);
                const float my = sy * (1.0f / 64.0f);
                // centered second moments
                float cxx = fmaf(-64.0f * mx, mx, sxx);
                float cyy = fmaf(-64.0f * my, my, syy);
                float cxy = fmaf(-64.0f * mx, my, sxy);
                cxx = fmaxf(cxx, 0.0f);
                cyy = fmaxf(cyy, 0.0f);
                // std with ddof=1, +eps (reference: (v-mean)/(std+eps))
                const float isx = 1.0f / (sqrtf(cxx * (1.0f / 63.0f)) + 1e-8f);
                const float isy = 1.0f / (sqrtf(cyy * (1.0f / 63.0f)) + 1e-8f);
                // sum (xhat - yhat)^2 = cxx*isx^2 + cyy*isy^2 - 2*cxy*isx*isy
                const float sq = cxx * isx * isx + cyy * isy * isy
                               - 2.0f * cxy * isx * isy;
                out[grow] = __expf(-sq) + cp * cp;   // GAMMA = 1
            }
        }
    }
}

extern "C" void kernel_launch(void* const* d_in, const int* in_sizes, int n_in,
                              void* d_out, int out_size, void* d_ws, size_t ws_size,
                              hipStream_t stream) {
    (void)n_in; (void)out_size; (void)d_ws; (void)ws_size;
    const float* x = (const float*)d_in[0];
    const float* y = (const float*)d_in[1];
    float* out = (float*)d_out;
    const int n_rows = in_sizes[0] / DIM;
    const int blocks = (n_rows + ROWS_PER_BLOCK - 1) / ROWS_PER_BLOCK;
    hipLaunchKernelGGL(hybrid_rbf_kernel, dim3(blocks), dim3(THREADS), 0, stream,
                       x, y, out, n_rows);
}